// WeightedCenterLoss_54116587929862
// MI455X (gfx1250) — compile-verified
//
#include <hip/hip_runtime.h>
#include <stdint.h>

#define NUM_CLASSES 100000
#define FEAT_DIM    512
#define BATCH       16384
#define ALPHA       0.5f
#define LAMDA       0.003f

#define D4 (FEAT_DIM / 4)                                   // 128 float4 per row
#define LOSS_ELEMS   ((size_t)BATCH * FEAT_DIM)             // 8,388,608
#define CENTER_ELEMS ((size_t)NUM_CLASSES * FEAT_DIM)       // 51,200,000

// The async-LDS builtins take (int4 AS1* global, int4 AS3* lds, imm off, imm cpol)
typedef int v4i __attribute__((ext_vector_type(4)));
typedef __attribute__((address_space(1))) v4i gas_v4i;      // global
typedef __attribute__((address_space(3))) v4i lds_v4i;      // LDS

__device__ __forceinline__ void wait_async0() {
#if __has_builtin(__builtin_amdgcn_s_wait_asynccnt)
    __builtin_amdgcn_s_wait_asynccnt(0);
#else
    asm volatile("s_wait_asynccnt 0" ::: "memory");
#endif
}

// ---------------- centers -> new_centers streaming copy via async LDS engine ----
// Unrolled x2: queue two async global->LDS loads (two 4KB tiles), one wait,
// queue two async LDS->global stores, one wait. 32B/lane per wait pair.
__global__ void k_copy_centers(const float* __restrict__ src, float* __restrict__ dst) {
    __shared__ __align__(16) float tile[2 * 256 * 4];       // two 4 KB staging tiles
    const unsigned N4 = (unsigned)(CENTER_ELEMS / 4);       // 12,800,000 float4
    const unsigned tid = threadIdx.x;
    const unsigned lds0 = (unsigned)(uintptr_t)tile + tid * 16u;
    const unsigned lds1 = lds0 + 4096u;
    const unsigned nthreads = gridDim.x * blockDim.x;
    const unsigned stride2 = 2u * nthreads;

    unsigned i = blockIdx.x * blockDim.x + tid;
    // main unrolled-by-2 loop
    for (; i + nthreads < N4; i += stride2) {
        const unsigned ia = i, ib = i + nthreads;
        __builtin_amdgcn_global_load_async_to_lds_b128(
            (gas_v4i*)(uintptr_t)(src + (size_t)ia * 4), (lds_v4i*)lds0, 0, 0);
        __builtin_amdgcn_global_load_async_to_lds_b128(
            (gas_v4i*)(uintptr_t)(src + (size_t)ib * 4), (lds_v4i*)lds1, 0, 0);
        wait_async0();                                      // both tiles filled
        __builtin_amdgcn_global_store_async_from_lds_b128(
            (gas_v4i*)(uintptr_t)(dst + (size_t)ia * 4), (lds_v4i*)lds0, 0, 0);
        __builtin_amdgcn_global_store_async_from_lds_b128(
            (gas_v4i*)(uintptr_t)(dst + (size_t)ib * 4), (lds_v4i*)lds1, 0, 0);
        wait_async0();                                      // tiles drained before reuse
    }
    // tail
    if (i < N4) {
        __builtin_amdgcn_global_load_async_to_lds_b128(
            (gas_v4i*)(uintptr_t)(src + (size_t)i * 4), (lds_v4i*)lds0, 0, 0);
        wait_async0();
        __builtin_amdgcn_global_store_async_from_lds_b128(
            (gas_v4i*)(uintptr_t)(dst + (size_t)i * 4), (lds_v4i*)lds0, 0, 0);
        wait_async0();
    }
}

// ---------------- counts ----------------
__global__ void k_zero_counts(unsigned* __restrict__ counts) {
    unsigned i = blockIdx.x * blockDim.x + threadIdx.x;
    if (i < NUM_CLASSES) counts[i] = 0u;
}

__global__ void k_count(const int* __restrict__ targets, unsigned* __restrict__ counts) {
    unsigned b = blockIdx.x * blockDim.x + threadIdx.x;
    if (b < BATCH) atomicAdd(&counts[targets[b]], 1u);
}

// ---------------- fused loss / weights / scatter center-update ----------------
// grid = BATCH blocks, block = 128 threads (4 wave32); one sample row per block,
// so targets[b]/counts[t]/wpc[t] are wave-uniform -> scalar loads.
__global__ void k_loss_scatter(const float4* __restrict__ x4,
                               const float* __restrict__ centers,
                               const float* __restrict__ wpc,
                               const int* __restrict__ targets,
                               const unsigned* __restrict__ counts,
                               float4* __restrict__ loss4,
                               float4* __restrict__ tw4,
                               float* __restrict__ out_centers) {
    const int b = blockIdx.x;
    const int j = threadIdx.x;                              // 0..127
    const int t = targets[b];
    const float w     = LAMDA * wpc[t];
    const float scale = -ALPHA / (1.0f + (float)counts[t]);

    const float4 x = x4[b * D4 + j];
    const float4 c = ((const float4*)(centers + (size_t)t * FEAT_DIM))[j];

    const float dx = x.x - c.x, dy = x.y - c.y, dz = x.z - c.z, dw = x.w - c.w;
    loss4[b * D4 + j] = make_float4(0.5f * dx * dx, 0.5f * dy * dy,
                                    0.5f * dz * dz, 0.5f * dw * dw);
    tw4[b * D4 + j] = make_float4(w, w, w, w);

    float* oc = out_centers + (size_t)t * FEAT_DIM + (size_t)j * 4;
    atomicAdd(oc + 0, scale * x.x);
    atomicAdd(oc + 1, scale * x.y);
    atomicAdd(oc + 2, scale * x.z);
    atomicAdd(oc + 3, scale * x.w);
}

extern "C" void kernel_launch(void* const* d_in, const int* in_sizes, int n_in,
                              void* d_out, int out_size, void* d_ws, size_t ws_size,
                              hipStream_t stream) {
    (void)in_sizes; (void)n_in; (void)out_size; (void)ws_size;

    const float* inputs  = (const float*)d_in[0];   // [16384, 512]
    const float* centers = (const float*)d_in[1];   // [100000, 512]
    const float* wpc     = (const float*)d_in[2];   // [100000]
    const int*   targets = (const int*)d_in[3];     // [16384]

    float* loss        = (float*)d_out;             // [16384, 512]
    float* tw          = loss + LOSS_ELEMS;         // [16384, 512]
    float* new_centers = tw + LOSS_ELEMS;           // [100000, 512]

    unsigned* counts = (unsigned*)d_ws;             // [100000] u32

    k_zero_counts<<<(NUM_CLASSES + 255) / 256, 256, 0, stream>>>(counts);
    k_count<<<(BATCH + 255) / 256, 256, 0, stream>>>(targets, counts);
    k_copy_centers<<<4096, 256, 0, stream>>>(centers, new_centers);
    k_loss_scatter<<<BATCH, D4, 0, stream>>>((const float4*)inputs, centers, wpc,
                                             targets, counts,
                                             (float4*)loss, (float4*)tw, new_centers);
}